// LmHeadWithPenaltySampleHead_39333310497460
// MI455X (gfx1250) — compile-verified
//
#include <hip/hip_runtime.h>
#include <hip/hip_bf16.h>
#include <math.h>

#define HDIM    4096
#define VOCAB   50257
#define LSEQ    2048
#define TOPK    50
#define MINKEEP 5
#define LN_EPS  1e-5f

#define COLS_PER_WAVE   32
#define WAVES_PER_BLOCK 8
#define NTG ((VOCAB + COLS_PER_WAVE - 1) / COLS_PER_WAVE)   // 1571 column groups

typedef __attribute__((ext_vector_type(2))) float v2f;
typedef __attribute__((ext_vector_type(8))) float v8f;

#define WMMA_F32_16x16x4(a, b, c) \
  __builtin_amdgcn_wmma_f32_16x16x4_f32(false, (a), false, (b), (short)0, (c), false, false)

// ---------------- Kernel 1: final LayerNorm over H=4096 ----------------
__global__ void ln_kernel(const float* __restrict__ x,
                          const float* __restrict__ gamma,
                          const float* __restrict__ beta,
                          float* __restrict__ xn) {
  __shared__ float s_sum[256];
  __shared__ float s_sq[256];
  int tid = threadIdx.x;
  float sum = 0.f, sq = 0.f;
  for (int i = tid; i < HDIM; i += 256) {
    float v = x[i];
    sum += v; sq += v * v;
  }
  s_sum[tid] = sum; s_sq[tid] = sq;
  __syncthreads();
  for (int s = 128; s > 0; s >>= 1) {
    if (tid < s) { s_sum[tid] += s_sum[tid + s]; s_sq[tid] += s_sq[tid + s]; }
    __syncthreads();
  }
  float mu   = s_sum[0] * (1.0f / HDIM);
  float var  = s_sq[0] * (1.0f / HDIM) - mu * mu;   // jnp.var (ddof=0)
  float rstd = rsqrtf(var + LN_EPS);
  for (int i = tid; i < HDIM; i += 256)
    xn[i] = (x[i] - mu) * rstd * gamma[i] + beta[i];
}

// ---------------- Kernel 2: GEMV [1,H] @ [H,V] via V_WMMA_F32_16X16X4_F32 ----
// One wave produces 32 vocab columns (two 16-wide WMMA tiles sharing the A
// fragment). B operand = 4x16 f32 tile of W, loaded as contiguous 64B rows per
// half-wave:
//   VGPR0: lane n (0-15) = W[k+0, col0+n], lane 16+n = W[k+2, col0+n]
//   VGPR1: lane n        = W[k+1, col0+n], lane 16+n = W[k+3, col0+n]
// A carries x in matrix row 0 only (lanes 0 and 16); result row M=0 of C is
// accumulator VGPR0, lanes 0-15.
__global__ void gemv_wmma_kernel(const float* __restrict__ xn,
                                 const float* __restrict__ W,
                                 float* __restrict__ logits) {
  __shared__ float sx[HDIM];
  int tid = threadIdx.x;
  for (int i = tid; i < HDIM; i += 256) sx[i] = xn[i];
  __syncthreads();

  const int wave = tid >> 5;
  const int lane = tid & 31;
  const int tg   = blockIdx.x * WAVES_PER_BLOCK + wave;
  if (tg >= NTG) return;                    // wave-uniform: EXEC stays all-1s

  const int col0 = tg * COLS_PER_WAVE;
  const int kh   = (lane >> 4) << 1;        // 0 for lanes 0-15, 2 for 16-31
  const int sub  = lane & 15;
  const bool row0 = (sub == 0);             // lanes 0/16 carry x (A row M=0)

  v8f c0 = {}, c1 = {};

  if (col0 + COLS_PER_WAVE <= VOCAB) {
    // ---- fast path: running pointer, second row/tile via immediate offsets
    const float* __restrict__ p = W + (size_t)kh * VOCAB + (col0 + sub);
    #pragma unroll 4
    for (int k = 0; k < HDIM; k += 4) {
      v2f a, b0, b1;
      float xa = sx[k + kh];                // K = kh      (broadcast LDS read)
      float xb = sx[k + kh + 1];            // K = kh + 1
      a.x = row0 ? xa : 0.0f;
      a.y = row0 ? xb : 0.0f;
      b0.x = p[0];                          // W[k+kh,   col0+sub]
      b0.y = p[VOCAB];                      // W[k+kh+1, col0+sub]
      b1.x = p[16];                         // second tile, +64B immediate
      b1.y = p[VOCAB + 16];
      c0 = WMMA_F32_16x16x4(a, b0, c0);
      c1 = WMMA_F32_16x16x4(a, b1, c1);
      p += 4 * VOCAB;
    }
  } else {
    // ---- edge path (last group only): clamp columns so no read passes V-1
    int colA = col0 + sub;       if (colA >= VOCAB) colA = VOCAB - 1;
    int colB = col0 + 16 + sub;  if (colB >= VOCAB) colB = VOCAB - 1;
    const float* __restrict__ pA = W + (size_t)kh * VOCAB + colA;
    const float* __restrict__ pB = W + (size_t)kh * VOCAB + colB;
    #pragma unroll 4
    for (int k = 0; k < HDIM; k += 4) {
      v2f a, b0, b1;
      float xa = sx[k + kh];
      float xb = sx[k + kh + 1];
      a.x = row0 ? xa : 0.0f;
      a.y = row0 ? xb : 0.0f;
      b0.x = pA[0];  b0.y = pA[VOCAB];
      b1.x = pB[0];  b1.y = pB[VOCAB];
      c0 = WMMA_F32_16x16x4(a, b0, c0);
      c1 = WMMA_F32_16x16x4(a, b1, c1);
      pA += 4 * VOCAB;
      pB += 4 * VOCAB;
    }
  }

  if (lane < 16) {                          // row M=0 of C = accumulator VGPR0
    int oc0 = col0 + lane;
    int oc1 = col0 + 16 + lane;
    if (oc0 < VOCAB) logits[oc0] = c0[0];
    if (oc1 < VOCAB) logits[oc1] = c1[0];
  }
}

// ---------------- Kernel 3: repetition penalty (gather -> rescale -> scatter) --
// Single block: read all originals first, barrier, then write, so every write is
// computed from the pre-penalty logits (matches JAX gather/scatter semantics;
// duplicate ids write identical values -> deterministic).
__global__ void penalty_kernel(const int* __restrict__ ids,
                               const float* __restrict__ penalty,
                               float* __restrict__ logits) {
  int tid = threadIdx.x;                    // 1024 threads, 2 tokens each
  float p = penalty[0];
  int id0 = ids[tid];
  int id1 = ids[tid + 1024];
  float g0 = logits[id0];
  float g1 = logits[id1];
  __syncthreads();
  logits[id0] = (g0 < 0.f) ? g0 * p : g0 / p;
  logits[id1] = (g1 < 0.f) ? g1 * p : g1 / p;
}

// ---------------- Kernel 4: top-k(50) + temperature + top-p + softmax ---------
__global__ void sample_kernel(float* __restrict__ logits,
                              const float* __restrict__ top_p,
                              const float* __restrict__ temperature,
                              float* __restrict__ out) {
  __shared__ float rv[1024];
  __shared__ int   ri[1024];
  __shared__ float topv[TOPK];
  __shared__ int   topi[TOPK];
  int tid = threadIdx.x;

  for (int j = 0; j < TOPK; ++j) {
    float bv = -INFINITY; int bi = 0x7fffffff;
    for (int i = tid; i < VOCAB; i += 1024) {
      float v = logits[i];
      if (v > bv) { bv = v; bi = i; }       // strict >: lower index wins in-thread
    }
    rv[tid] = bv; ri[tid] = bi;
    __syncthreads();
    for (int s = 512; s > 0; s >>= 1) {
      if (tid < s) {
        float ov = rv[tid + s]; int oi = ri[tid + s];
        if (ov > rv[tid] || (ov == rv[tid] && oi < ri[tid])) { rv[tid] = ov; ri[tid] = oi; }
      }
      __syncthreads();
    }
    if (tid == 0) {
      topv[j] = rv[0]; topi[j] = ri[0];
      logits[ri[0]] = -INFINITY;            // scratch copy, regenerated each call
    }
    __syncthreads();
  }

  if (tid == 0) {
    float invT = 1.0f / temperature[0];
    float P    = top_p[0];
    float l[TOPK], e[TOPK], f[TOPK];
    for (int j = 0; j < TOPK; ++j) l[j] = topv[j] * invT;
    // softmax over sorted-desc logits for the cumulative sum
    float m = l[0], sum = 0.f;
    for (int j = 0; j < TOPK; ++j) { e[j] = expf(l[j] - m); sum += e[j]; }
    float inv = 1.0f / sum, cum = 0.f;
    for (int j = 0; j < TOPK; ++j) {
      cum += e[j] * inv;
      bool keep = (cum < P) || (j < MINKEEP);
      f[j] = keep ? l[j] : -1000.0f;
    }
    // final softmax over filtered logits
    float m2 = f[0];
    for (int j = 1; j < TOPK; ++j) m2 = fmaxf(m2, f[j]);
    float s2 = 0.f;
    for (int j = 0; j < TOPK; ++j) { f[j] = expf(f[j] - m2); s2 += f[j]; }
    float inv2 = 1.0f / s2;
    for (int j = 0; j < TOPK; ++j) out[j]        = f[j] * inv2;
    for (int j = 0; j < TOPK; ++j) out[TOPK + j] = (float)topi[j];
  }
}

extern "C" void kernel_launch(void* const* d_in, const int* in_sizes, int n_in,
                              void* d_out, int out_size, void* d_ws, size_t ws_size,
                              hipStream_t stream) {
  const float* hs    = (const float*)d_in[0];   // hidden_states [1,4096]
  const int*   ids   = (const int*)  d_in[1];   // input_ids     [1,2048]
  const float* top_p = (const float*)d_in[2];   // [1,1]
  const float* temp  = (const float*)d_in[3];   // [1,1]
  const float* pen   = (const float*)d_in[4];   // [1,1]
  const float* gamma = (const float*)d_in[5];   // [4096]
  const float* beta  = (const float*)d_in[6];   // [4096]
  const float* W     = (const float*)d_in[7];   // [4096,50257]
  float* out = (float*)d_out;                   // 100: probs[50] ++ tokens[50]

  float* xn     = (float*)d_ws;                 // [4096]
  float* logits = xn + HDIM;                    // [50257]

  ln_kernel<<<1, 256, 0, stream>>>(hs, gamma, beta, xn);

  int nblocks = (NTG + WAVES_PER_BLOCK - 1) / WAVES_PER_BLOCK;
  gemv_wmma_kernel<<<nblocks, 256, 0, stream>>>(xn, W, logits);

  penalty_kernel<<<1, 1024, 0, stream>>>(ids, pen, logits);

  sample_kernel<<<1, 1024, 0, stream>>>(logits, top_p, temp, out);
}